// STLVQE_LUT_21036749816345
// MI455X (gfx1250) — compile-verified
//
#include <hip/hip_runtime.h>
#include <stdint.h>

#define LL 128
#define THDV 127.0f

// Fixed problem shape from the reference (B,1,H,W), makes /W,/H strength-reduce.
constexpr int Bc = 4;
constexpr int Hc = 540;
constexpr int Wc = 960;
constexpr int HWc = Hc * Wc;          // 518400 (divisible by 4)
constexpr int Nc  = Bc * HWc;         // 2073600
constexpr int W4  = Wc / 4;           // 240 groups per row
constexpr int P4  = Nc / 4;           // 518400 4-pixel groups
constexpr int NTHR = 1024;            // 32 wave32s per block

// ---------------------------------------------------------------------------
// Branchless triangular 2-D LUT interpolation, accumulating 4 channels.
//   fa>=fb : (1-fa)*v00 + (fa-fb)*v(ia1,ib) + fb*v11
//   else   : (1-fb)*v00 + (fb-fa)*v(ia,ib1) + fa*v11
//   => w00 = 1-max(fa,fb), wm = |fa-fb|, w11 = min(fa,fb), select mid vertex.
// 3x ds_load_b128 from the LDS-resident (Tri-permuted) LUT per call.
// ---------------------------------------------------------------------------
__device__ __forceinline__ void tri_interp_acc(float a, float b,
                                               const float4* __restrict__ sLUT,
                                               float& ax, float& ay,
                                               float& az, float& aw)
{
    float fia = floorf(a);
    float fib = floorf(b);
    int   ia  = (int)fia;
    int   ib  = (int)fib;
    float fa  = a - fia;
    float fb  = b - fib;
    int   ia1 = min(ia + 1, LL - 1);
    int   ib1 = min(ib + 1, LL - 1);

    float4 v00 = sLUT[ia * LL + ib];
    float4 v11 = sLUT[ia1 * LL + ib1];
    int    im  = (fa >= fb) ? (ia1 * LL + ib) : (ia * LL + ib1);
    float4 vm  = sLUT[im];

    float w00 = 1.0f - fmaxf(fa, fb);
    float wm  = fabsf(fa - fb);
    float w11 = fminf(fa, fb);

    ax += w00 * v00.x + wm * vm.x + w11 * v11.x;
    ay += w00 * v00.y + wm * vm.y + w11 * v11.y;
    az += w00 * v00.z + wm * vm.z + w11 * v11.z;
    aw += w00 * v00.w + wm * vm.w + w11 * v11.w;
}

__device__ __forceinline__ float clip_q(float v, float inv_s)
{
    return fminf(fmaxf(v * inv_s, 0.0f), THDV);
}

__global__ __launch_bounds__(NTHR)
void stlvqe_lut_kernel(const float* __restrict__ x,
                       const float* __restrict__ s_df,
                       const float* __restrict__ LUT,
                       const int*   __restrict__ tri,
                       float*       __restrict__ out)
{
    // 256 KB: entire Tri-permuted LUT in LDS (320 KB per WGP on CDNA5).
    __shared__ float4 sLUT[LL * LL];

    const int tid = threadIdx.x;

    // -----------------------------------------------------------------------
    // Stage sLUT[k] = LUT[tri[k]] with CDNA5 async global->LDS gathers.
    // Load 16 tri indices per thread as four b128 loads (in flight together),
    // then issue 16 GLOBAL_LOAD_ASYNC_TO_LDS_B128, one ASYNCcnt wait total.
    // -----------------------------------------------------------------------
    {
        const uint32_t lds_base = (uint32_t)(uintptr_t)(&sLUT[0]);
        const int4* __restrict__ tri4 = (const int4*)tri;

        int4 t0 = tri4[tid];
        int4 t1 = tri4[tid + NTHR];
        int4 t2 = tri4[tid + 2 * NTHR];
        int4 t3 = tri4[tid + 3 * NTHR];

#define ASYNC_GATHER(T, M, C)                                                 \
        do {                                                                  \
            uint64_t ga = (uint64_t)(uintptr_t)(LUT + (size_t)(T) * 4);       \
            uint32_t la = lds_base + (uint32_t)(M) * 64u + (C) * 16u;         \
            asm volatile("global_load_async_to_lds_b128 %0, %1, off"          \
                         :: "v"(la), "v"(ga) : "memory");                     \
        } while (0)

        ASYNC_GATHER(t0.x, tid,            0); ASYNC_GATHER(t0.y, tid,            1);
        ASYNC_GATHER(t0.z, tid,            2); ASYNC_GATHER(t0.w, tid,            3);
        ASYNC_GATHER(t1.x, tid + NTHR,     0); ASYNC_GATHER(t1.y, tid + NTHR,     1);
        ASYNC_GATHER(t1.z, tid + NTHR,     2); ASYNC_GATHER(t1.w, tid + NTHR,     3);
        ASYNC_GATHER(t2.x, tid + 2*NTHR,   0); ASYNC_GATHER(t2.y, tid + 2*NTHR,   1);
        ASYNC_GATHER(t2.z, tid + 2*NTHR,   2); ASYNC_GATHER(t2.w, tid + 2*NTHR,   3);
        ASYNC_GATHER(t3.x, tid + 3*NTHR,   0); ASYNC_GATHER(t3.y, tid + 3*NTHR,   1);
        ASYNC_GATHER(t3.z, tid + 3*NTHR,   2); ASYNC_GATHER(t3.w, tid + 3*NTHR,   3);
#undef ASYNC_GATHER

        asm volatile("s_wait_asynccnt 0" ::: "memory");
    }
    __syncthreads();

    const float s     = s_df[0];      // s_scale == s in forward value
    const float inv_s = 1.0f / s;     // one divide per thread, 5 muls/pixel saved

    for (int g = blockIdx.x * NTHR + tid; g < P4; g += gridDim.x * NTHR) {
        int t    = g / W4;            // global row index = b*H + i
        int j0   = (g - t * W4) << 2; // first column of this 4-pixel group
        int i    = t % Hc;
        int bimg = t / Hc;
        int idx  = t * Wc + j0;       // flat pixel index of pixel 0

        // Center row (aligned b128), up/down rows (aligned, edge-clamped),
        // and the two edge scalars for left/right neighbors.
        const float4 xc4 = *(const float4*)(x + idx);
        const int offu = (i > 0)      ? -Wc : 0;
        const int offd = (i < Hc - 1) ?  Wc : 0;
        const float4 xu4 = *(const float4*)(x + idx + offu);
        const float4 xd4 = *(const float4*)(x + idx + offd);
        const float  xlo = x[idx - (j0 > 0 ? 1 : 0)];
        const float  xhi = x[idx + 3 + (j0 + 4 < Wc ? 1 : 0)];

        float qc[4] = { clip_q(xc4.x, inv_s), clip_q(xc4.y, inv_s),
                        clip_q(xc4.z, inv_s), clip_q(xc4.w, inv_s) };
        float qU[4] = { clip_q(xu4.x, inv_s), clip_q(xu4.y, inv_s),
                        clip_q(xu4.z, inv_s), clip_q(xu4.w, inv_s) };
        float qD[4] = { clip_q(xd4.x, inv_s), clip_q(xd4.y, inv_s),
                        clip_q(xd4.z, inv_s), clip_q(xd4.w, inv_s) };
        const float qlo = clip_q(xlo, inv_s);
        const float qhi = clip_q(xhi, inv_s);
        // Left/right neighbor q reuse the center vector (edge-clamped).
        float qL[4] = { qlo,   qc[0], qc[1], qc[2] };
        float qR[4] = { qc[1], qc[2], qc[3], qhi  };

        float ax[4] = {0.f,0.f,0.f,0.f}, ay[4] = {0.f,0.f,0.f,0.f};
        float az[4] = {0.f,0.f,0.f,0.f}, aw[4] = {0.f,0.f,0.f,0.f};
#pragma unroll
        for (int k = 0; k < 4; ++k) {
            tri_interp_acc(qc[k], qR[k], sLUT, ax[k], ay[k], az[k], aw[k]); // r=0 right
            tri_interp_acc(qc[k], qD[k], sLUT, ax[k], ay[k], az[k], aw[k]); // r=1 down
            tri_interp_acc(qc[k], qL[k], sLUT, ax[k], ay[k], az[k], aw[k]); // r=2 left
            tri_interp_acc(qc[k], qU[k], sLUT, ax[k], ay[k], az[k], aw[k]); // r=3 up
        }

        // out[b, c, i, j] flat = idx + 3*b*HW + c*HW ; 16B-aligned b128 stores.
        const int base = idx + 3 * bimg * HWc;
        *(float4*)(out + base)           = make_float4(0.25f*ax[0], 0.25f*ax[1], 0.25f*ax[2], 0.25f*ax[3]);
        *(float4*)(out + base + HWc)     = make_float4(0.25f*ay[0], 0.25f*ay[1], 0.25f*ay[2], 0.25f*ay[3]);
        *(float4*)(out + base + 2*HWc)   = make_float4(0.25f*az[0], 0.25f*az[1], 0.25f*az[2], 0.25f*az[3]);
        *(float4*)(out + base + 3*HWc)   = make_float4(0.25f*aw[0], 0.25f*aw[1], 0.25f*aw[2], 0.25f*aw[3]);
    }
}

extern "C" void kernel_launch(void* const* d_in, const int* in_sizes, int n_in,
                              void* d_out, int out_size, void* d_ws, size_t ws_size,
                              hipStream_t stream)
{
    const float* x   = (const float*)d_in[0];
    const float* sdf = (const float*)d_in[1];
    const float* lut = (const float*)d_in[2];
    const int*   tri = (const int*)d_in[3];
    float*       out = (float*)d_out;

    // 256 blocks x 1024 threads: ~2 four-pixel groups per thread; 256KB LDS
    // caps at 1 block/WGP, 32 wave32s per WGP for latency hiding.
    dim3 block(NTHR);
    dim3 grid(256);
    hipLaunchKernelGGL(stlvqe_lut_kernel, grid, block, 0, stream,
                       x, sdf, lut, tri, out);
    (void)d_ws; (void)ws_size; (void)in_sizes; (void)n_in; (void)out_size;
}